// Extractor_10806137717387
// MI455X (gfx1250) — compile-verified
//
#include <hip/hip_runtime.h>
#include <cstdint>

// ---------------------------------------------------------------------------
// Trilinear TSDF fusion sampling for MI455X (gfx1250).
// Memory-bound: ~376MB NT-streamed output; both 64MB volumes kept resident in
// the 192MB L2 by using non-temporal hints on all streaming traffic.
// No matmul structure -> no WMMA; CDNA5 paths used: async global->LDS DMA
// (ASYNCcnt + s_wait_asynccnt), ds loads, NT (TH_NT) loads/stores,
// global_prefetch. Gathers are branchless (clamped index * mask, matching the
// reference semantics) to avoid per-corner EXEC save/restore under wave32.
// ---------------------------------------------------------------------------

typedef float f32x4 __attribute__((ext_vector_type(4)));
typedef int   i32x4 __attribute__((ext_vector_type(4)));

// Builtin prototype (confirmed via hipcc diagnostic):
//   void __builtin_amdgcn_global_load_async_to_lds_b32(
//       __device__ int* gsrc, __shared__ int* ldst, imm int offset, imm int cpol)
typedef __attribute__((address_space(1))) int* gptr_i32;
typedef __attribute__((address_space(3))) int* lptr_i32;

#if defined(__AMDGCN__) && __has_builtin(__builtin_amdgcn_global_load_async_to_lds_b32)
#define ASYNC_LDS 1
#else
#define ASYNC_LDS 0
#endif

template <bool ALIGNED>
__global__ __launch_bounds__(256) void trilerp_kernel(
    const float* __restrict__ points,
    const float* __restrict__ tsdf,
    const float* __restrict__ wvol,
    float* __restrict__ out,
    const long long M, const int D)
{
  const int t = (int)threadIdx.x;
  const long long base = (long long)blockIdx.x * 256ll;
  const long long i = base + t;

  float px, py, pz;

#if ASYNC_LDS
  // Stage this block's 256 points (768 floats) coalesced into LDS with the
  // CDNA5 async DMA path, then read per-lane (stride-3 -> bank-conflict-free).
  __shared__ float sp[768];
  {
    const long long g0 = base * 3ll;
    const float* gp = points + g0 + t;
    if (base + 256ll <= M) {
      // Full block: uniform branch, no EXEC manipulation on the DMA issues.
      __builtin_amdgcn_global_load_async_to_lds_b32(
          (gptr_i32)(gp + 0),   (lptr_i32)(&sp[t + 0]),   0, 0);
      __builtin_amdgcn_global_load_async_to_lds_b32(
          (gptr_i32)(gp + 256), (lptr_i32)(&sp[t + 256]), 0, 0);
      __builtin_amdgcn_global_load_async_to_lds_b32(
          (gptr_i32)(gp + 512), (lptr_i32)(&sp[t + 512]), 0, 0);
    } else {
      const long long lim3 = M * 3ll;
#pragma unroll
      for (int ch = 0; ch < 3; ++ch) {
        if (g0 + (long long)(ch * 256 + t) < lim3) {
          __builtin_amdgcn_global_load_async_to_lds_b32(
              (gptr_i32)(gp + ch * 256), (lptr_i32)(&sp[ch * 256 + t]), 0, 0);
        }
      }
    }
  }
#if __has_builtin(__builtin_amdgcn_s_wait_asynccnt)
  __builtin_amdgcn_s_wait_asynccnt(0);
#else
  asm volatile("s_wait_asynccnt 0" ::: "memory");
#endif
  __syncthreads();
  px = sp[3 * t + 0];
  py = sp[3 * t + 1];
  pz = sp[3 * t + 2];
  if (i >= M) return;
#else
  if (i >= M) return;
  // Streaming, used-once data: non-temporal loads keep L2 free for volumes.
  px = __builtin_nontemporal_load(points + 3 * i + 0);
  py = __builtin_nontemporal_load(points + 3 * i + 1);
  pz = __builtin_nontemporal_load(points + 3 * i + 2);
#endif

  // Hint the next block's point tile toward the caches (global_prefetch_b8).
  if (i + 256 < M) __builtin_prefetch(points + 3 * (i + 256), 0, 0);

  const float p3[3] = {px, py, pz};
  float fl[3], a[3], ai[3], nb[3];
#pragma unroll
  for (int d = 0; d < 3; ++d) {
    const float f = floorf(p3[d]);           // idx
    const float diff = (f + 0.5f) - p3[d];   // center - p
    nb[d] = (diff > 0.0f ? 1.0f : 0.0f) - (diff < 0.0f ? 1.0f : 0.0f); // sign
    const float al = fabsf(diff);            // alpha = |p - center|
    fl[d] = f;
    a[d]  = al;
    ai[d] = 1.0f - al;
  }

  int   ci[24];
  float w8[8];
  float fv = 0.0f, fw = 0.0f;
  const int Dm1 = D - 1;
#pragma unroll
  for (int c = 0; c < 8; ++c) {
    const int ox = (c >> 2) & 1, oy = (c >> 1) & 1, oz = c & 1;
    const float w = (ox ? a[0] : ai[0]) * (oy ? a[1] : ai[1]) * (oz ? a[2] : ai[2]);
    const int cx = (int)(fl[0] + (ox ? nb[0] : 0.0f));
    const int cy = (int)(fl[1] + (oy ? nb[1] : 0.0f));
    const int cz = (int)(fl[2] + (oz ? nb[2] : 0.0f));
    ci[3 * c + 0] = cx;
    ci[3 * c + 1] = cy;
    ci[3 * c + 2] = cz;
    w8[c] = w;
    // Branchless: gather at clamped index, zero via mask (reference semantics).
    const int kx = min(max(cx, 0), Dm1);
    const int ky = min(max(cy, 0), Dm1);
    const int kz = min(max(cz, 0), Dm1);
    const int lin = (kx * D + ky) * D + kz;   // gathers: default RT hint, L2-resident
    const bool valid = ((unsigned)cx < (unsigned)D) &
                       ((unsigned)cy < (unsigned)D) &
                       ((unsigned)cz < (unsigned)D);
    const float wm = valid ? w : 0.0f;
    fv = fmaf(tsdf[lin], wm, fv);
    fw = fmaf(wvol[lin], wm, fw);
  }

  // Output layout (concatenated, in reference return order):
  //   [0,M)       fusion_values   f32
  //   [M,25M)     indices         i32  (M*8*3)
  //   [25M,33M)   weights         f32  (M*8)
  //   [33M,34M)   fusion_weights  f32
  float* __restrict__ out_fv = out;
  int*   __restrict__ out_ix = (int*)(out + M);
  float* __restrict__ out_w  = out + 25ll * M;
  float* __restrict__ out_fw = out + 33ll * M;

  __builtin_nontemporal_store(fv, out_fv + i);
  __builtin_nontemporal_store(fw, out_fw + i);

  if (ALIGNED) {
    // 16B-aligned b128 NT stores: 6x int4 + 2x float4 per point.
    i32x4* ip = (i32x4*)(out_ix + 24ll * i);
#pragma unroll
    for (int k = 0; k < 6; ++k) {
      const i32x4 v = {ci[4 * k + 0], ci[4 * k + 1], ci[4 * k + 2], ci[4 * k + 3]};
      __builtin_nontemporal_store(v, ip + k);
    }
    f32x4* wp = (f32x4*)(out_w + 8ll * i);
    const f32x4 w0 = {w8[0], w8[1], w8[2], w8[3]};
    const f32x4 w1 = {w8[4], w8[5], w8[6], w8[7]};
    __builtin_nontemporal_store(w0, wp + 0);
    __builtin_nontemporal_store(w1, wp + 1);
  } else {
#pragma unroll
    for (int k = 0; k < 24; ++k)
      __builtin_nontemporal_store(ci[k], out_ix + 24ll * i + k);
#pragma unroll
    for (int c = 0; c < 8; ++c)
      __builtin_nontemporal_store(w8[c], out_w + 8ll * i + c);
  }
}

extern "C" void kernel_launch(void* const* d_in, const int* in_sizes, int n_in,
                              void* d_out, int out_size, void* d_ws, size_t ws_size,
                              hipStream_t stream) {
  (void)n_in; (void)out_size; (void)d_ws; (void)ws_size;
  const float* points = (const float*)d_in[0];
  const float* tsdf   = (const float*)d_in[1];
  const float* wvol   = (const float*)d_in[2];
  float* out = (float*)d_out;

  const long long M = (long long)in_sizes[0] / 3ll;   // total sample points

  // Volume is cubic: recover D from element count (integer cube root).
  const long long vol = (long long)in_sizes[1];
  int D = 1;
  while ((long long)(D + 1) * (D + 1) * (D + 1) <= vol) ++D;

  const int blocks = (int)((M + 255ll) / 256ll);
  const bool aligned = ((M & 3ll) == 0ll) && ((((uintptr_t)d_out) & 15u) == 0u);

  if (aligned)
    trilerp_kernel<true><<<blocks, 256, 0, stream>>>(points, tsdf, wvol, out, M, D);
  else
    trilerp_kernel<false><<<blocks, 256, 0, stream>>>(points, tsdf, wvol, out, M, D);
}